// TotalLoss_80032420594333
// MI455X (gfx1250) — compile-verified
//
#include <hip/hip_runtime.h>

typedef float v2f __attribute__((ext_vector_type(2)));
typedef float v8f __attribute__((ext_vector_type(8)));

#define NBATCH 2048
#define NC 128
#define GH 200
#define GW 200

// loss-term weights (means folded to constants)
#define WHOL  (500.0f / (2048.0f * 127.0f))
#define WCUR  (500.0f / (2048.0f * 127.0f))
#define WTURN (100.0f / (2048.0f * 126.0f))
#define WSAFE (500.0f / (2048.0f * 256.0f))
#define WARC  (0.5f   / 2048.0f)
#define WUNI  (100.0f / 2048.0f)
#define WTRAJ (0.3f   / 2048.0f)
#define KMAX2 (1.67f * 1.67f)

__device__ __forceinline__ float psl1(float x) {
  const float pe = 1.0e-4f;               // f3c = 1/pe^2 = 1e8, f4c = -0.5*f3c/pe = -5e11
  if (x >= pe)   return x - 0.5f * pe;
  if (x > 0.0f)  return (-5.0e11f * x + 1.0e8f) * x * x * x;
  return 0.0f;
}

__device__ __forceinline__ const float* esdf_cell(const float* env, float px, float py) {
  float cx = fminf(fmaxf(px, -9.9f), 9.9f);
  float cy = fminf(fmaxf(py, -9.9f), 9.9f);
  int ix = (int)floorf((cx - 0.05f + 10.0f) * 10.0f);
  int iy = (int)floorf((cy - 0.05f + 10.0f) * 10.0f);
  return env + ix * GW + iy;
}

__device__ __forceinline__ float safety_term(const float* env, float px, float py) {
  bool outr = (px < -9.9f) || (px > 9.9f) || (py < -9.9f) || (py > 9.9f);
  float cx = fminf(fmaxf(px, -9.9f), 9.9f);
  float cy = fminf(fmaxf(py, -9.9f), 9.9f);
  int ix = (int)floorf((cx - 0.05f + 10.0f) * 10.0f);
  int iy = (int)floorf((cy - 0.05f + 10.0f) * 10.0f);
  float ixp = ((float)ix + 0.5f) * 0.1f - 10.0f;
  float iyp = ((float)iy + 0.5f) * 0.1f - 10.0f;
  float dxw = (cx - ixp) * 10.0f;
  float dyw = (cy - iyp) * 10.0f;
  const float* r0 = env + ix * GW + iy;
  float g00 = r0[0];
  float g01 = r0[1];
  float g10 = r0[GW];
  float g11 = r0[GW + 1];
  float v00 = (1.0f - dxw) * g00 + dxw * g10;
  float v10 = (1.0f - dxw) * g01 + dxw * g11;
  float d = (1.0f - dyw) * v00 + dyw * v10;
  if (outr) d = -1.0f;
  return psl1(10.0f * (0.3f - d));
}

// One workgroup (4 wave32s) per batch. Produces one partial-loss float per batch.
__global__ __launch_bounds__(NC) void loss_batch_kernel(
    const float* __restrict__ opState, const float* __restrict__ labState,
    const float* __restrict__ opRot,   const float* __restrict__ labRot,
    const float* __restrict__ envs,    float* __restrict__ partial)
{
  const int b = blockIdx.x;
  const int t = threadIdx.x;

  __shared__ float2 sOp[NC], sLab[NC], sRot[NC], sLRot[NC];
  __shared__ float2 sV[NC - 1], sVL[NC - 1];
  __shared__ float  red[8][NC];

  const float* env = envs + (size_t)b * (GH * GW);
  const size_t pbase = ((size_t)b * NC + t) * 2;

  float2 op  = make_float2(opState[pbase],  opState[pbase + 1]);
  float2 lb  = make_float2(labState[pbase], labState[pbase + 1]);
  float2 rt  = make_float2(opRot[pbase],    opRot[pbase + 1]);
  float2 lrt = make_float2(labRot[pbase],   labRot[pbase + 1]);

  // Contact points: offset = R(rot) @ (px, 0) = (cos*px, sin*px); rot stored raw as (cos,sin).
  float a0x = op.x + rt.x * 0.15f, a0y = op.y + rt.y * 0.15f;
  float a1x = op.x + rt.x * 0.45f, a1y = op.y + rt.y * 0.45f;

  // Early prefetch of the scattered ESDF cells (global_prefetch_b8) to overlap the barrier.
  __builtin_prefetch(esdf_cell(env, a0x, a0y), 0, 1);
  __builtin_prefetch(esdf_cell(env, a1x, a1y), 0, 1);

  sOp[t] = op; sLab[t] = lb; sRot[t] = rt; sLRot[t] = lrt;
  __syncthreads();

  float acc = 0.0f;
  acc += WSAFE * (safety_term(env, a0x, a0y) + safety_term(env, a1x, a1y));

  float segLen = 0.0f, labLen = 0.0f, rotLen = 0.0f, lrotLen = 0.0f, seg2 = 0.0f;
  if (t < NC - 1) {
    float dx = sOp[t + 1].x - sOp[t].x, dy = sOp[t + 1].y - sOp[t].y;
    float d2 = dx * dx + dy * dy;
    segLen = sqrtf(d2);
    seg2 = d2;
    float inv = 1.0f / fmaxf(segLen, 1e-12f);
    float vx = dx * inv, vy = dy * inv;
    sV[t] = make_float2(vx, vy);

    float lx = sLab[t + 1].x - sLab[t].x, ly = sLab[t + 1].y - sLab[t].y;
    labLen = sqrtf(lx * lx + ly * ly);
    float linv = 1.0f / fmaxf(labLen, 1e-12f);
    sVL[t] = make_float2(lx * linv, ly * linv);

    float rx = sRot[t + 1].x - sRot[t].x, ry = sRot[t + 1].y - sRot[t].y;
    float r2 = rx * rx + ry * ry;
    rotLen = sqrtf(r2);

    float qx = sLRot[t + 1].x - sLRot[t].x, qy = sLRot[t + 1].y - sLRot[t].y;
    lrotLen = sqrtf(qx * qx + qy * qy);

    // holonomic: cross(normalized opState segment, raw opRot[c+1])
    float labx = sRot[t + 1].x, laby = sRot[t + 1].y;
    float cross = vx * (-laby) + vy * labx;
    acc += WHOL * psl1(cross * cross - 0.067f);

    // curvature: d_ang^2 - kmax^2 * (d_arc + 1e-3)^2
    float da = segLen + 0.001f;
    acc += WCUR * psl1(r2 - KMAX2 * da * da);
  }
  __syncthreads();

  float stO = 0.0f, stL = 0.0f;
  if (t < NC - 2) {
    float2 v1 = sV[t], v2 = sV[t + 1];
    float crs = v1.x * (-v2.y) + v1.y * v2.x;
    float c2 = crs * crs;
    stO = c2;
    acc += WTURN * psl1(c2 - 0.35f);
    float2 w1 = sVL[t], w2 = sVL[t + 1];
    float crl = w1.x * (-w2.y) + w1.y * w2.x;
    stL = crl * crl;
  }

  red[0][t] = acc;
  red[1][t] = segLen;   // arc_loss(opState)       and S1 for std
  red[2][t] = labLen;   // arc_loss(label_opState)
  red[3][t] = rotLen;   // arc_loss(opRot)
  red[4][t] = lrotLen;  // arc_loss(label_opRot)
  red[5][t] = seg2;     // S2 for std
  red[6][t] = stO;      // smooth_traj(opState)
  red[7][t] = stL;      // smooth_traj(label_opState)
  __syncthreads();

  for (int s = NC / 2; s > 0; s >>= 1) {
    if (t < s) {
#pragma unroll
      for (int k = 0; k < 8; ++k) red[k][t] += red[k][t + s];
    }
    __syncthreads();
  }

  if (t == 0) {
    float arcO = red[1][0], arcL = red[2][0], arcR = red[3][0], arcLR = red[4][0];
    float S2 = red[5][0], sto = red[6][0], stl = red[7][0];
    float mean = arcO * (1.0f / 127.0f);
    float var  = fmaxf(S2 * (1.0f / 127.0f) - mean * mean, 0.0f);
    float res = red[0][0]
              + WARC  * psl1(arcO - arcL)          // arcloss
              + WARC  * psl1(arcR - arcLR)         // rsmloss
              + WUNI  * sqrtf(var) / arcL          // unilos
              + WTRAJ * psl1(sto - stl);           // trajloss
    partial[b] = res;
  }
}

// Final 2048 -> 1 reduction using V_WMMA_F32_16X16X4_F32 with an all-ones B
// matrix: D = A*1 + C accumulates row sums of 64 partials per issue, full f32.
// One wave32, EXEC all ones (WMMA requirement).
__global__ __launch_bounds__(32) void reduce_wmma_kernel(
    const float* __restrict__ p, float* __restrict__ out, int n)
{
  const int lane = threadIdx.x;
  const float2* p2 = (const float2*)p;

  v8f c = {0.0f, 0.0f, 0.0f, 0.0f, 0.0f, 0.0f, 0.0f, 0.0f};
  v2f bones; bones.x = 1.0f; bones.y = 1.0f;

  const int iters = n >> 6;  // 64 values per WMMA
  for (int it = 0; it < iters; ++it) {
    float2 aa = p2[it * 32 + lane];
    v2f a; a.x = aa.x; a.y = aa.y;
    // (neg_a, A, neg_b, B, c_mod, C, reuse_a, reuse_b)
    c = __builtin_amdgcn_wmma_f32_16x16x4_f32(
        false, a, false, bones, (short)0, c, false, false);
  }

  // Each D column is identical (B == ones): sum of all 256 entries = 16 * total.
  float s = c[0] + c[1] + c[2] + c[3] + c[4] + c[5] + c[6] + c[7];
  for (int off = 16; off > 0; off >>= 1)
    s += __shfl_down(s, off, 32);
  if (lane == 0) out[0] = s * (1.0f / 16.0f);
}

extern "C" void kernel_launch(void* const* d_in, const int* in_sizes, int n_in,
                              void* d_out, int out_size, void* d_ws, size_t ws_size,
                              hipStream_t stream) {
  const float* opState  = (const float*)d_in[0];
  const float* labState = (const float*)d_in[1];
  const float* opRot    = (const float*)d_in[2];
  const float* labRot   = (const float*)d_in[3];
  const float* envs     = (const float*)d_in[4];
  float* partial = (float*)d_ws;                 // 2048 floats of scratch
  float* out     = (float*)d_out;

  loss_batch_kernel<<<NBATCH, NC, 0, stream>>>(opState, labState, opRot, labRot,
                                               envs, partial);
  reduce_wmma_kernel<<<1, 32, 0, stream>>>(partial, out, NBATCH);
}